// RNNLayer_87153476371020
// MI455X (gfx1250) — compile-verified
//
#include <hip/hip_runtime.h>
#include <math.h>

// CDNA5 (gfx1250) wave32: V_WMMA_F32_16X16X4_F32 + Tensor Data Mover staging.
typedef __attribute__((ext_vector_type(2)))  float        v2f;
typedef __attribute__((ext_vector_type(8)))  float        v8f;
typedef __attribute__((ext_vector_type(4)))  unsigned int u32x4;
typedef __attribute__((ext_vector_type(4)))  int          i32x4;
typedef __attribute__((ext_vector_type(8)))  int          i32x8;

#define SEQ   2048
#define BATCH 32
#define EMB   512
#define HID   512
// TDM pad: 1 DWORD inserted per 256 DWORDs stored -> LDS row stride 514 floats.
// 514 mod 64banks = 2, so the 16 A-fragment lanes (row stride) hit 16 banks.
#define APAD  514
#define LDS_BYTES (32 * APAD * 4)   // 65792 B (dynamic LDS; WGP has 320 KB)

#if __has_builtin(__builtin_amdgcn_tensor_load_to_lds)
#define HAVE_TDM 1
#else
#define HAVE_TDM 0
#endif

// element (m, k) of a TDM-padded 512-wide row in LDS
__device__ __forceinline__ int lds_off(int m, int k) {
    return m * APAD + k + (k >> 8);   // +1 pad dword once k >= 256
}

#if HAVE_TDM
__device__ __forceinline__ void tdm_load(u32x4 g0, i32x8 g1, i32x4 g2, i32x4 g3) {
#if defined(__clang_major__) && (__clang_major__ >= 23)
    i32x8 g4 = {};
    __builtin_amdgcn_tensor_load_to_lds(g0, g1, g2, g3, g4, 0);
#else
    __builtin_amdgcn_tensor_load_to_lds(g0, g1, g2, g3, 0);
#endif
}

// D# group 0: count=1, optional gather mode (16-bit indices), lds/global addr, type=2
__device__ __forceinline__ u32x4 make_g0(unsigned lds_addr, const void* gptr, int gather) {
    unsigned long long ga = (unsigned long long)(uintptr_t)gptr;
    u32x4 g = {};
    g[0] = 1u | ((unsigned)gather << 31);              // count=1, gather_mode, 16b idx
    g[1] = lds_addr;                                   // LDS byte address
    g[2] = (unsigned)(ga & 0xFFFFFFFFull);             // global_addr[31:0]
    g[3] = (unsigned)((ga >> 32) & 0x01FFFFFFull) | (2u << 30);  // [56:32], type=2
    return g;
}

// D# group 1: data_size=4B, pad 1 DW per 256 DW, dims/strides (data_size units)
__device__ __forceinline__ i32x8 make_g1(int tdim0, int tdim1, int tile0, int tile1,
                                         int stride0) {
    i32x8 g = {};
    g[0] = (2 << 16) | (1 << 20) | (7 << 22);          // data_size=2(4B), pad_en, ivl=256
    g[1] = (tdim0 & 0xFFFF) << 16;                     // tensor_dim0[15:0]
    g[2] = ((tdim0 >> 16) & 0xFFFF) | ((tdim1 & 0xFFFF) << 16);
    g[3] = ((tdim1 >> 16) & 0xFFFF) | ((tile0 & 0xFFFF) << 16);
    g[4] = (tile1 & 0xFFFF);                           // tile_dim1 (gather: #indices)
    g[5] = stride0;                                    // tensor_dim0_stride[31:0]
    return g;
}
#endif

// -------------------------------------------------------------------------
// Phase 1: embedding gather + input projection (M=T*B=65536, N=K=512)
//   xp[m][n] = sum_k emb[src[m]][k] * W_ih[n][k] + b_ih[n]
// Block 256 thr (8 waves), tile 32M x 64N, one 16x16 C tile per wave.
// A-block (32 gathered rows x 512) staged by two TDM gather-mode descriptors
// (16-bit row indices = tokens), padded for conflict-free fragment reads.
// -------------------------------------------------------------------------
__global__ __launch_bounds__(256) void embed_proj_kernel(
    const int*   __restrict__ src,
    const float* __restrict__ emb,
    const float* __restrict__ W_ih,
    const float* __restrict__ b_ih,
    float*       __restrict__ xp)
{
    extern __shared__ float As[];        // 32 x APAD (TDM-padded rows)

    const int tid  = threadIdx.x;
    const int lane = tid & 31;
    const int wave = tid >> 5;
    const int mtile = wave & 1;
    const int ntile = wave >> 1;
    const int mblock = blockIdx.x * 32;
    const int nbase  = blockIdx.y * 64 + ntile * 16;

    const int hi2  = lane >> 4;
    const int l15  = lane & 15;
    const int arow = mtile * 16 + l15;
    const int ncol = nbase + l15;

#if HAVE_TDM
    if (wave == 0) {
        int tk[32];
        #pragma unroll
        for (int i = 0; i < 32; ++i) tk[i] = src[mblock + i];
        i32x8 g1 = make_g1(EMB, 32000, EMB, /*#indices*/16, EMB);
        i32x4 g2, g3;
        #pragma unroll
        for (int j = 0; j < 4; ++j) {   // rows 0..15
            g2[j] = (tk[2*j]     & 0xFFFF) | ((tk[2*j + 1]  & 0xFFFF) << 16);
            g3[j] = (tk[8 + 2*j] & 0xFFFF) | ((tk[9 + 2*j]  & 0xFFFF) << 16);
        }
        tdm_load(make_g0(0, emb, 1), g1, g2, g3);
        #pragma unroll
        for (int j = 0; j < 4; ++j) {   // rows 16..31
            g2[j] = (tk[16 + 2*j] & 0xFFFF) | ((tk[17 + 2*j] & 0xFFFF) << 16);
            g3[j] = (tk[24 + 2*j] & 0xFFFF) | ((tk[25 + 2*j] & 0xFFFF) << 16);
        }
        tdm_load(make_g0(16 * APAD * 4, emb, 1), g1, g2, g3);
        __builtin_amdgcn_s_wait_tensorcnt(0);
    }
#else
    for (int idx = tid; idx < 32 * EMB; idx += 256) {
        int m = idx >> 9, k = idx & 511;
        As[lds_off(m, k)] = emb[(size_t)src[mblock + m] * EMB + k];
    }
#endif
    __syncthreads();

    v8f c = {};
    const float* wrow = W_ih + (size_t)ncol * EMB + hi2 * 2;

    #pragma unroll 8
    for (int k4 = 0; k4 < EMB; k4 += 4) {
        int kk = k4 + hi2 * 2;
        int ko = lds_off(arow, kk);
        v2f a;  a.x = As[ko];  a.y = As[ko + 1];
        v2f b = *(const v2f*)(wrow + k4);
        c = __builtin_amdgcn_wmma_f32_16x16x4_f32(
                false, a, false, b, (short)0, c, false, false);
    }

    const float bias = b_ih[ncol];
    #pragma unroll
    for (int i = 0; i < 8; ++i) {
        int m = mblock + mtile * 16 + i + hi2 * 8;
        xp[(size_t)m * HID + ncol] = c[i] + bias;
    }
}

// -------------------------------------------------------------------------
// Phase 2: one recurrence step; launch boundary = grid-wide sync.
//   h_out[m][n] = tanh(xp[m][n] + sum_k h_prev[m][k]*W_hh[n][k] + b_hh[n])
// h_prev (32x512) staged into padded LDS by ONE tensor_load_to_lds.
// W_hh (1 MB) stays resident in the 192 MB L2 across all 2048 steps.
// -------------------------------------------------------------------------
__global__ __launch_bounds__(256) void rnn_step_kernel(
    const float* __restrict__ W_hh,
    const float* __restrict__ b_hh,
    const float* __restrict__ h_prev,
    float*       __restrict__ h_out,
    int has_prev)
{
    extern __shared__ float hS[];        // 32 x APAD

    const int tid  = threadIdx.x;
    const int lane = tid & 31;
    const int wave = tid >> 5;
    const int mtile = wave & 1;
    const int ntile = wave >> 1;
    const int nbase = blockIdx.x * 64 + ntile * 16;

    const int hi2  = lane >> 4;
    const int l15  = lane & 15;
    const int arow = mtile * 16 + l15;
    const int ncol = nbase + l15;

    if (has_prev) {
#if HAVE_TDM
        if (wave == 0) {
            i32x4 z = {};
            tdm_load(make_g0(0, h_prev, 0),
                     make_g1(HID, BATCH, HID, BATCH, HID), z, z);
            __builtin_amdgcn_s_wait_tensorcnt(0);
        }
#else
        for (int idx = tid; idx < 32 * HID; idx += 256) {
            int m = idx >> 9, k = idx & 511;
            hS[lds_off(m, k)] = h_prev[(size_t)m * HID + k];
        }
#endif
    } else {
        for (int idx = tid; idx < 32 * APAD; idx += 256) hS[idx] = 0.0f;
    }
    __syncthreads();

    v8f c = {};
    const float* wrow = W_hh + (size_t)ncol * HID + hi2 * 2;

    #pragma unroll 8
    for (int k4 = 0; k4 < HID; k4 += 4) {
        if ((k4 & 127) == 0)
            __builtin_prefetch(wrow + k4 + 128, 0, 0);
        int kk = k4 + hi2 * 2;
        int ko = lds_off(arow, kk);
        v2f a;  a.x = hS[ko];  a.y = hS[ko + 1];
        v2f b = *(const v2f*)(wrow + k4);
        c = __builtin_amdgcn_wmma_f32_16x16x4_f32(
                false, a, false, b, (short)0, c, false, false);
    }

    const float bias = b_hh[ncol];
    #pragma unroll
    for (int i = 0; i < 8; ++i) {
        int m = mtile * 16 + i + hi2 * 8;
        size_t off = (size_t)m * HID + ncol;
        h_out[off] = tanhf(c[i] + h_out[off] + bias);   // in-place over x_proj[t]
    }
}

// -------------------------------------------------------------------------
// d_in: 0 src(int32), 1 lengths(unused), 2 emb_table, 3 W_ih, 4 W_hh,
//       5 b_ih, 6 b_hh.   d_out: [layer_final | outputs] fp32.
// -------------------------------------------------------------------------
extern "C" void kernel_launch(void* const* d_in, const int* in_sizes, int n_in,
                              void* d_out, int out_size, void* d_ws, size_t ws_size,
                              hipStream_t stream)
{
    const int*   src  = (const int*)d_in[0];
    const float* emb  = (const float*)d_in[2];
    const float* W_ih = (const float*)d_in[3];
    const float* W_hh = (const float*)d_in[4];
    const float* b_ih = (const float*)d_in[5];
    const float* b_hh = (const float*)d_in[6];

    float* out     = (float*)d_out;
    float* outputs = out + (size_t)BATCH * HID;   // outputs region doubles as x_proj

    dim3 gA(SEQ * BATCH / 32, HID / 64);
    embed_proj_kernel<<<gA, 256, LDS_BYTES, stream>>>(src, emb, W_ih, b_ih, outputs);

    for (int t = 0; t < SEQ; ++t) {
        const float* hp = (t == 0) ? outputs
                                   : outputs + (size_t)(t - 1) * BATCH * HID;
        rnn_step_kernel<<<dim3(HID / 64), 256, LDS_BYTES, stream>>>(
            W_hh, b_hh, hp, outputs + (size_t)t * BATCH * HID, t > 0 ? 1 : 0);
    }

    hipMemcpyAsync(out, outputs + (size_t)(SEQ - 1) * BATCH * HID,
                   (size_t)BATCH * HID * sizeof(float),
                   hipMemcpyDeviceToDevice, stream);
}